// AngularAEVComputer_44092134260987
// MI455X (gfx1250) — compile-verified
//
#include <hip/hip_runtime.h>

typedef float v2f __attribute__((ext_vector_type(2)));
typedef float v4f __attribute__((ext_vector_type(4)));
typedef float v8f __attribute__((ext_vector_type(8)));

#define RCA_F 3.5f
#define NPAIRS 10
#define NQ 32
#define NBATCH 16
#define NATOM 64
#define TMAX 1968   /* ceil16(63*62/2 = 1953) */

__constant__ float c_SHFZ[8] = {0.19634954f, 0.58904862f, 0.9817477f, 1.3744468f,
                                1.7671459f, 2.1598449f, 2.552544f, 2.9452431f};
__constant__ float c_SHFA[4] = {0.9f, 1.55f, 2.2f, 2.85f};

__global__ __launch_bounds__(128)
void AngularAEV_wmma_kernel(const float* __restrict__ dmat,
                            const int* __restrict__ species,
                            float* __restrict__ out)
{
    __shared__ float sR[NATOM];
    __shared__ float sFc[NATOM];
    __shared__ int   sJ[NATOM];
    __shared__ int   sS[NATOM];
    __shared__ int   sCnt;
    __shared__ float sAcc[NPAIRS * NQ];
    __shared__ v4f   sTrip[TMAX];   // {cosA, sinA, mR, fcp} per triple
    __shared__ int   sPidx[TMAX];   // phase2: packed (a<<8|bb); phase3: pidx

    const int bi   = blockIdx.x;          // center (b,i)
    const int b    = bi >> 6;
    const int i    = bi & 63;
    const int tid  = threadIdx.x;
    const int lane = tid & 31;
    const int wave = tid >> 5;
    const int hi   = lane >> 4;           // half-wave: triples {0,1} vs {2,3}
    const int q0   = lane & 15;           // q within 16-tile == e column for C

    for (int idx = tid; idx < NPAIRS * NQ; idx += 128) sAcc[idx] = 0.0f;
    if (tid == 0) sCnt = 0;
    __syncthreads();

    // ---- phase 1: compact neighbor list for center (b,i) ----
    const float* drow = dmat + (size_t)b * (NATOM * NATOM) + (size_t)i * NATOM;
    if (tid < NATOM) {
        float dij = drow[tid];
        if (dij < RCA_F && dij != 0.0f) {
            int slot = atomicAdd(&sCnt, 1);
            sR[slot]  = dij;
            sFc[slot] = 0.5f * __cosf((3.14159265358979f / RCA_F) * dij) + 0.5f;
            sJ[slot]  = tid;
            sS[slot]  = species[b * NATOM + tid];
        }
    }
    __syncthreads();

    const int M = sCnt;
    const int T = (M * (M - 1)) >> 1;
    const int Tpad = (T + 15) & ~15;

    // ---- phase 2: pair table, integer only (parallel over rows a) ----
    for (int a = tid; a < M - 1; a += 128) {
        int t = (a * (2 * M - a - 1)) >> 1;
        for (int bb = a + 1; bb < M; ++bb) sPidx[t++] = (a << 8) | bb;
    }
    __syncthreads();

    // ---- phase 3: per-triple scalars, computed ONCE per triple ----
    for (int t = tid; t < Tpad; t += 128) {
        v4f tr = {0.f, 0.f, 0.f, 0.f};
        int pidx = 0;
        if (t < T) {
            const int pr = sPidx[t];
            const int a  = pr >> 8;
            const int bb = pr & 255;
            const float Rij = sR[a];
            const float Rik = sR[bb];
            const float Rjk = dmat[(size_t)b * (NATOM * NATOM) +
                                   (size_t)sJ[a] * NATOM + sJ[bb]];
            const float fcp = sFc[a] * sFc[bb];
            const float denom = fmaxf(2.0f * Rij * Rik, 1e-10f);
            const float cos_a = (Rij * Rij + Rik * Rik - Rjk * Rjk) *
                                __builtin_amdgcn_rcpf(denom);
            const float cA = 0.95f * cos_a;                       // cos(alpha)
            const float sA = __builtin_amdgcn_sqrtf(fmaxf(1.0f - cA * cA, 0.0f));
            tr[0] = cA; tr[1] = sA; tr[2] = 0.5f * (Rij + Rik); tr[3] = fcp;
            const int s1 = sS[a], s2 = sS[bb];
            const int x = (s1 < s2 ? s1 : s2) - 1;
            const int y = (s1 < s2 ? s2 : s1) - 1;
            pidx = x * (9 - x) / 2 + (y - x);
        }
        sTrip[t] = tr;
        sPidx[t] = pidx;
    }
    __syncthreads();

    // ---- per-lane constants ----
    const float shz   = c_SHFZ[lane & 7];
    const float cz    = __cosf(shz);
    const float szv   = __sinf(shz);
    const float sa_lo = c_SHFA[q0 >> 3];
    const float sa_hi = c_SHFA[(q0 >> 3) + 2];

    v8f c0 = {0.f,0.f,0.f,0.f,0.f,0.f,0.f,0.f};   // D[q 0..15][e]
    v8f c1 = {0.f,0.f,0.f,0.f,0.f,0.f,0.f,0.f};   // D[q 16..31][e]

    // ---- phase 4: branchless WMMA main loop ----
    for (int base = wave * 4; base < T; base += 16) {
        const int tA = base + (hi << 1);
        const v4f trA = sTrip[tA];
        const v4f trB = sTrip[tA + 1];
        const int pA  = sPidx[tA];
        const int pB  = sPidx[tA + 1];

        // triple A
        float cc = trA[0] * cz + trA[1] * szv;    // cos(alpha - Sz)
        float u  = 0.5f + 0.5f * cc;
        u *= u; u *= u; u *= u; u *= u; u *= u;   // u^32
        float g  = 2.0f * u * trA[3];
        float d0 = trA[2] - sa_lo;
        float d1 = trA[2] - sa_hi;
        const float gLoA = g * __expf(-8.0f * d0 * d0);
        const float gHiA = g * __expf(-8.0f * d1 * d1);

        // triple B
        cc = trB[0] * cz + trB[1] * szv;
        u  = 0.5f + 0.5f * cc;
        u *= u; u *= u; u *= u; u *= u; u *= u;
        g  = 2.0f * u * trB[3];
        d0 = trB[2] - sa_lo;
        d1 = trB[2] - sa_hi;
        const float gLoB = g * __expf(-8.0f * d0 * d0);
        const float gHiB = g * __expf(-8.0f * d1 * d1);

        // A (16x4 f32): VGPR0 = K0(lanes0-15)/K2(lanes16-31), VGPR1 = K1/K3
        v2f Alo; Alo[0] = gLoA; Alo[1] = gLoB;
        v2f Ahi; Ahi[0] = gHiA; Ahi[1] = gHiB;
        // B (4x16 f32 one-hot over pair-channel e)
        v2f Bm;  Bm[0] = (pA == q0) ? 1.0f : 0.0f;
                 Bm[1] = (pB == q0) ? 1.0f : 0.0f;

        c0 = __builtin_amdgcn_wmma_f32_16x16x4_f32(false, Alo, false, Bm, (short)0, c0, false, false);
        c1 = __builtin_amdgcn_wmma_f32_16x16x4_f32(false, Ahi, false, Bm, (short)0, c1, false, false);
    }

    // ---- merge wave-private C tiles: C[r]: M=r+8*hi (q), N=q0 (e) ----
    if (q0 < NPAIRS) {
        #pragma unroll
        for (int r = 0; r < 8; ++r) {
            const int qs = r + (hi << 3);
            atomicAdd(&sAcc[q0 * NQ + qs],      c0[r]);
            atomicAdd(&sAcc[q0 * NQ + 16 + qs], c1[r]);
        }
    }
    __syncthreads();

    float* orow = out + (size_t)bi * (NPAIRS * NQ);
    for (int idx = tid; idx < NPAIRS * NQ; idx += 128) orow[idx] = sAcc[idx];
}

extern "C" void kernel_launch(void* const* d_in, const int* in_sizes, int n_in,
                              void* d_out, int out_size, void* d_ws, size_t ws_size,
                              hipStream_t stream) {
    const float* dmat    = (const float*)d_in[0];   // (16,64,64) f32
    const int*   species = (const int*)d_in[1];     // (16,64) i32
    float*       out     = (float*)d_out;           // (1024,10,32) f32
    (void)in_sizes; (void)n_in; (void)out_size; (void)d_ws; (void)ws_size;
    AngularAEV_wmma_kernel<<<NBATCH * NATOM, 128, 0, stream>>>(dmat, species, out);
}